// Evolution_11441792876696
// MI455X (gfx1250) — compile-verified
//
#include <hip/hip_runtime.h>
#include <hip/hip_bf16.h>
#include <math.h>
#include <stdint.h>

typedef float v2f __attribute__((ext_vector_type(2)));
typedef float v8f __attribute__((ext_vector_type(8)));
typedef unsigned int u32x4 __attribute__((ext_vector_type(4)));
typedef int i32x4 __attribute__((ext_vector_type(4)));
typedef int i32x8 __attribute__((ext_vector_type(8)));

#define N_ITERS 3
#define N_BLOCKS 3
#define HEADS 8
#define HDIM 16
#define DMODEL 128
#define FFDIM 1024
#define CSEG 32
#define NPT 20
#define NPOLY 512
#define MROWS (NPOLY * NPT)   /* 10240 */
#define HH 320
#define WW 320
#define EPSV 1e-5f
#define CLIP_DIS_V 100.0f
#define KC 64                 /* K-chunk staged in LDS per block */

#define WMMA_F32(a, b, c) \
    __builtin_amdgcn_wmma_f32_16x16x4_f32(false, (a), false, (b), (short)0, (c), false, false)

#if defined(__HIP_DEVICE_COMPILE__) && __has_builtin(__builtin_amdgcn_tensor_load_to_lds)
#define HAVE_TDM 1
#else
#define HAVE_TDM 0
#endif

#if __has_include(<hip/amd_detail/amd_gfx1250_TDM.h>)
#define TDM_SIX_ARGS 1
#else
#define TDM_SIX_ARGS 0
#endif

// Low 32 bits of a generic pointer to LDS == byte offset in the wave's LDS.
__device__ __forceinline__ unsigned lds_offset_of(const void* p)
{
    return (unsigned)(unsigned long long)(uintptr_t)p;
}

#if HAVE_TDM
// TDM 2-D tile load: rows x kc fp32 elements, global row stride strideElems.
// LDS destination gets 1 DWORD of padding after every kc DWORDs (bank-conflict
// avoidance: row stride kc+1), via D# pad_enable/pad_interval/pad_amount.
// Descriptor per CDNA5 ISA 8.3/8.4; groups 2/3 zero (2-D tile).
__device__ __forceinline__ void tdm_load_2d(unsigned ldsOff, const void* gptr,
                                            int kc, int rows, int strideElems)
{
    unsigned long long ga = (unsigned long long)(uintptr_t)gptr;
    u32x4 g0;
    g0[0] = 1u;                                   // count=1 (valid), no gather
    g0[1] = ldsOff;                               // lds_addr (bytes)
    g0[2] = (unsigned)(ga & 0xFFFFFFFFu);         // global_addr[31:0]
    g0[3] = (unsigned)((ga >> 32) & 0x01FFFFFFu)  // global_addr[56:32]
          | (2u << 30);                           // type = 2 ("image")
    const unsigned pic = (kc == 64) ? 5u : 4u;    // pad every 64 / 32 DWORDs
    unsigned td0 = (unsigned)kc;                  // tensor_dim0 == tile width
    unsigned td1 = (unsigned)rows;                // tensor_dim1 == tile rows
    unsigned long long s0 = (unsigned long long)(unsigned)strideElems;
    i32x8 g1;
    g1[0] = (int)((2u << 16)                      // data_size=2 -> 4 bytes
                | (1u << 20)                      // pad_enable
                | (pic << 22));                   // pad_interval; pad_amount=0 (1 DW)
    g1[1] = (int)((td0 & 0xFFFFu) << 16);         // tensor_dim0[15:0]
    g1[2] = (int)(((td0 >> 16) & 0xFFFFu) | ((td1 & 0xFFFFu) << 16));
    g1[3] = (int)(((td1 >> 16) & 0xFFFFu) | ((unsigned)kc << 16)); // tile_dim0
    g1[4] = (int)(td1 & 0xFFFFu);                 // tile_dim1=rows, tile_dim2=0
    g1[5] = (int)(s0 & 0xFFFFFFFFu);              // tensor_dim0_stride[31:0]
    g1[6] = (int)((s0 >> 32) & 0xFFFFu);          // stride[47:32]; dim1_stride=0
    g1[7] = 0;
    i32x4 z4 = {0, 0, 0, 0};
#if TDM_SIX_ARGS
    i32x8 z8 = {0, 0, 0, 0, 0, 0, 0, 0};
    __builtin_amdgcn_tensor_load_to_lds(g0, g1, z4, z4, z8, 0);
#else
    __builtin_amdgcn_tensor_load_to_lds(g0, g1, z4, z4, 0);
#endif
}
#endif

// One k-step of the macro-tile: 2 A fragments (global), 4 B fragments (LDS,
// row stride RS = kc+1 for bank-conflict-free access), 8 WMMAs.
#define GEMM_KSTEP(RS)                                                         \
    {                                                                          \
        const int ka = k0 + 2 * hi;                                            \
        const int kg = kbase + ka;                                             \
        v2f a0, a1, b0, b1, b2, b3;                                            \
        a0.x = xr0[kg]; a0.y = xr0[kg + 1];                                    \
        a1.x = xr1[kg]; a1.y = xr1[kg + 1];                                    \
        b0.x = buf[(ln)      * (RS) + ka]; b0.y = buf[(ln)      * (RS) + ka + 1]; \
        b1.x = buf[(16 + ln) * (RS) + ka]; b1.y = buf[(16 + ln) * (RS) + ka + 1]; \
        b2.x = buf[(32 + ln) * (RS) + ka]; b2.y = buf[(32 + ln) * (RS) + ka + 1]; \
        b3.x = buf[(48 + ln) * (RS) + ka]; b3.y = buf[(48 + ln) * (RS) + ka + 1]; \
        c[0][0] = WMMA_F32(a0, b0, c[0][0]);                                   \
        c[0][1] = WMMA_F32(a0, b1, c[0][1]);                                   \
        c[0][2] = WMMA_F32(a0, b2, c[0][2]);                                   \
        c[0][3] = WMMA_F32(a0, b3, c[0][3]);                                   \
        c[1][0] = WMMA_F32(a1, b0, c[1][0]);                                   \
        c[1][1] = WMMA_F32(a1, b1, c[1][1]);                                   \
        c[1][2] = WMMA_F32(a1, b2, c[1][2]);                                   \
        c[1][3] = WMMA_F32(a1, b3, c[1][3]);                                   \
    }

// ---------------------------------------------------------------------------
// WMMA fp32 GEMM, 32x64 macro-tile per wave (TM=2 x TN=4 of 16x16 WMMA tiles).
//   Y = act( (X @ W^T + bias) * (col<scaleLim ? scale : 1) + add1 + add2 )
// The 64-row W strip shared by the block's 4 waves is staged into LDS by the
// Tensor Data Mover in KC-wide chunks (double-buffered: DMA of chunk i+1
// overlaps WMMA on chunk i), rows padded to kc+1 DWORDs (conflict-free banks).
// Grouped mode (xSlab/ySlab != 0): column-third t reads X + t*xSlab, writes
// Y + t*ySlab (block-diagonal X) -- fuses the qq/kk/vv projection.
// ---------------------------------------------------------------------------
__global__ __launch_bounds__(128)
void gemm_wmma_f32(const float* __restrict__ X, const float* __restrict__ W,
                   const float* __restrict__ bias, const float* __restrict__ add1,
                   const float* __restrict__ add2, float* __restrict__ Y,
                   int M, int N, int K, int ldY, float scale, int scaleLim,
                   int relu, long xSlab, long ySlab)
{
    __shared__ float sW[2][64 * (KC + 1)];        // 2 x 16.25 KB, padded rows

    const int lane = threadIdx.x;                 // 0..31
    const int ln   = lane & 15;
    const int hi   = lane >> 4;                   // ISA: lanes 16-31 hold K+2,K+3
    const int nBase = blockIdx.x * 64;
    const int mBase = (blockIdx.y * 4 + threadIdx.y) * 32;

    const int third = nBase >> 7;                 // 128-col group (grouped mode)
    const float* Xb = xSlab ? X + (size_t)third * xSlab : X;
    float*       Yb = ySlab ? Y + (size_t)third * ySlab : Y;
    const int colLocalBase = ySlab ? (nBase & 127) : nBase;

    const float* xr0 = Xb + (size_t)(mBase + ln) * K;
    const float* xr1 = Xb + (size_t)(mBase + 16 + ln) * K;
    const float* gW  = W + (size_t)nBase * K;     // 64-row strip, row stride K

    __builtin_prefetch(xr0, 0, 3);
    __builtin_prefetch(xr1, 0, 3);

    const int kc = (K < KC) ? K : KC;             // 32 or 64; K % kc == 0
    const int rs = kc + 1;                        // padded LDS row stride
    const int tid = threadIdx.y * 32 + threadIdx.x;

    v8f c[2][4];
#pragma unroll
    for (int i = 0; i < 2; ++i)
#pragma unroll
        for (int j = 0; j < 4; ++j) c[i][j] = (v8f){};

    for (int ch = 0, kbase = 0; kbase < K; ++ch, kbase += kc) {
#if HAVE_TDM
        if (ch == 0) {
            if (threadIdx.y == 0) {
                tdm_load_2d(lds_offset_of(&sW[0][0]), gW, kc, 64, K);
                __builtin_amdgcn_s_wait_tensorcnt(0);
            }
            __syncthreads();
        }
        const int hasNext = (kbase + kc < K);
        if (hasNext && threadIdx.y == 0) {        // overlap DMA with compute
            tdm_load_2d(lds_offset_of(&sW[(ch + 1) & 1][0]),
                        gW + (size_t)(kbase + kc), kc, 64, K);
        }
        const float* buf = &sW[ch & 1][0];
#else
        __syncthreads();                          // cooperative fallback stage
        {
            const int ksh = (kc == 64) ? 6 : 5;
            const float* gsrc = gW + (size_t)kbase;
            for (int i = tid; i < 64 * kc; i += 128) {
                int row = i >> ksh, col = i & (kc - 1);
                sW[0][row * rs + col] = gsrc[(size_t)row * K + col];
            }
        }
        __syncthreads();
        const float* buf = &sW[0][0];
#endif
        if (kc == KC) {
#pragma unroll 4
            for (int k0 = 0; k0 < KC; k0 += 4) GEMM_KSTEP(KC + 1)
        } else {
            for (int k0 = 0; k0 < kc; k0 += 4) GEMM_KSTEP(rs)
        }
#if HAVE_TDM
        if (hasNext) {
            if (threadIdx.y == 0) __builtin_amdgcn_s_wait_tensorcnt(0);
            __syncthreads();                      // publish next buffer / reuse
        }
#endif
    }

    if (mBase >= M) return;                       // (grid is exact; safety)
#pragma unroll
    for (int i = 0; i < 2; ++i) {
#pragma unroll
        for (int j = 0; j < 4; ++j) {
            const int colG = nBase + j * 16 + ln;          // global col
            const int colL = colLocalBase + j * 16 + ln;   // col within Yb
            const float sc = (colG < scaleLim) ? scale : 1.0f;
            const float bv = bias ? bias[colG] : 0.0f;
#pragma unroll
            for (int r = 0; r < 8; ++r) {
                const int row = mBase + 16 * i + r + 8 * hi;
                float v = (c[i][j][r] + bv) * sc;
                const size_t idx = (size_t)row * ldY + colL;
                if (add1) v += add1[idx];
                if (add2) v += add2[idx];
                if (relu) v = fmaxf(v, 0.0f);
                Yb[idx] = v;
            }
        }
    }
}

// ---------------------------------------------------------------------------
// Bilinear gather of node features: one thread per (poly, point) row.
// ---------------------------------------------------------------------------
__global__ void gather_nf(const float* __restrict__ feat,
                          const float* __restrict__ polys,
                          const int* __restrict__ inds,
                          float* __restrict__ nf)
{
    int t = blockIdx.x * blockDim.x + threadIdx.x;
    if (t >= MROWS) return;
    int n = t / NPT;
    float x = polys[t * 2 + 0], y = polys[t * 2 + 1];
    float px = x * (float)(WW - 1) / (float)WW;
    float py = y * (float)(HH - 1) / (float)HH;
    float x0f = floorf(px), y0f = floorf(py);
    float wx = px - x0f, wy = py - y0f;
    int x0 = (int)fminf(fmaxf(x0f,        0.f), (float)(WW - 1));
    int x1 = (int)fminf(fmaxf(x0f + 1.f,  0.f), (float)(WW - 1));
    int y0 = (int)fminf(fmaxf(y0f,        0.f), (float)(HH - 1));
    int y1 = (int)fminf(fmaxf(y0f + 1.f,  0.f), (float)(HH - 1));
    int b = inds[n];
    float w00 = (1.f - wy) * (1.f - wx), w01 = (1.f - wy) * wx;
    float w10 = wy * (1.f - wx),         w11 = wy * wx;
    const float* base = feat + (size_t)b * CSEG * HH * WW;
#pragma unroll
    for (int c = 0; c < CSEG; ++c) {
        const float* fc = base + (size_t)c * HH * WW;
        float v = fc[y0 * WW + x0] * w00 + fc[y0 * WW + x1] * w01 +
                  fc[y1 * WW + x0] * w10 + fc[y1 * WW + x1] * w11;
        nf[(size_t)t * CSEG + c] = v;
    }
}

// Per-channel mean / inv-std over all MROWS rows (batch-norm stats).
__global__ __launch_bounds__(256)
void channel_stats(const float* __restrict__ nf, float* __restrict__ stats)
{
    __shared__ float ss[256], sq[256];
    int c = blockIdx.x;
    float s = 0.f, q = 0.f;
    for (int r = threadIdx.x; r < MROWS; r += 256) {
        float v = nf[(size_t)r * CSEG + c];
        s += v; q += v * v;
    }
    ss[threadIdx.x] = s; sq[threadIdx.x] = q;
    __syncthreads();
    for (int off = 128; off > 0; off >>= 1) {
        if (threadIdx.x < off) {
            ss[threadIdx.x] += ss[threadIdx.x + off];
            sq[threadIdx.x] += sq[threadIdx.x + off];
        }
        __syncthreads();
    }
    if (threadIdx.x == 0) {
        float mean = ss[0] / (float)MROWS;
        float var  = sq[0] / (float)MROWS - mean * mean;
        stats[c]        = mean;
        stats[CSEG + c] = rsqrtf(var + EPSV);
    }
}

__global__ void bn_apply(float* __restrict__ nf, const float* __restrict__ stats)
{
    int t = blockIdx.x * blockDim.x + threadIdx.x;
    if (t >= MROWS * CSEG) return;
    int c = t % CSEG;
    nf[t] = (nf[t] - stats[c]) * stats[CSEG + c];
}

// LayerNorm over last dim (128); one wave32 per row, 4 elements per lane.
__global__ __launch_bounds__(256)
void layernorm_rows(const float* __restrict__ X, const float* __restrict__ g,
                    const float* __restrict__ b, float* __restrict__ Y)
{
    int warp = threadIdx.x >> 5;
    int lane = threadIdx.x & 31;
    int row = blockIdx.x * 8 + warp;
    if (row >= MROWS) return;
    const float* x = X + (size_t)row * DMODEL;
    float v0 = x[lane], v1 = x[lane + 32], v2 = x[lane + 64], v3 = x[lane + 96];
    float s = v0 + v1 + v2 + v3;
#pragma unroll
    for (int off = 16; off > 0; off >>= 1) s += __shfl_xor(s, off, 32);
    float mean = s * (1.0f / DMODEL);
    float d0 = v0 - mean, d1 = v1 - mean, d2 = v2 - mean, d3 = v3 - mean;
    float q = d0 * d0 + d1 * d1 + d2 * d2 + d3 * d3;
#pragma unroll
    for (int off = 16; off > 0; off >>= 1) q += __shfl_xor(q, off, 32);
    float inv = rsqrtf(q * (1.0f / DMODEL) + EPSV);
    float* y = Y + (size_t)row * DMODEL;
    y[lane]      = d0 * inv * g[lane]      + b[lane];
    y[lane + 32] = d1 * inv * g[lane + 32] + b[lane + 32];
    y[lane + 64] = d2 * inv * g[lane + 64] + b[lane + 64];
    y[lane + 96] = d3 * inv * g[lane + 96] + b[lane + 96];
}

// Multi-head attention, seq len 20, 8 heads x 16 dim: one thread per (head, q).
__global__ __launch_bounds__(192)
void attention_kernel(const float* __restrict__ qq, const float* __restrict__ kk,
                      const float* __restrict__ vv, float* __restrict__ ao)
{
    int n = blockIdx.x;
    int t = threadIdx.x;
    if (t >= HEADS * NPT) return;
    int h = t / NPT, qi = t % NPT;
    const float* qr = qq + ((size_t)(n * NPT + qi)) * DMODEL + h * HDIM;
    float qv[HDIM];
#pragma unroll
    for (int d = 0; d < HDIM; ++d) qv[d] = qr[d];
    float sc[NPT];
    float mx = -1e30f;
#pragma unroll
    for (int j = 0; j < NPT; ++j) {
        const float* kr = kk + ((size_t)(n * NPT + j)) * DMODEL + h * HDIM;
        float acc = 0.f;
#pragma unroll
        for (int d = 0; d < HDIM; ++d) acc += qv[d] * kr[d];
        sc[j] = acc; mx = fmaxf(mx, acc);
    }
    float sum = 0.f;
#pragma unroll
    for (int j = 0; j < NPT; ++j) { sc[j] = expf(sc[j] - mx); sum += sc[j]; }
    float inv = 1.0f / sum;
    float out[HDIM];
#pragma unroll
    for (int d = 0; d < HDIM; ++d) out[d] = 0.f;
#pragma unroll
    for (int j = 0; j < NPT; ++j) {
        const float* vr = vv + ((size_t)(n * NPT + j)) * DMODEL + h * HDIM;
        float a = sc[j] * inv;
#pragma unroll
        for (int d = 0; d < HDIM; ++d) out[d] += a * vr[d];
    }
    float* orow = ao + ((size_t)(n * NPT + qi)) * DMODEL + h * HDIM;
#pragma unroll
    for (int d = 0; d < HDIM; ++d) orow[d] = out[d];
}

__global__ void copy_flat(const float* __restrict__ src, float* __restrict__ dst, int n)
{
    int t = blockIdx.x * blockDim.x + threadIdx.x;
    if (t < n) dst[t] = src[t];
}

// Copy hdn (ld 128) into xcat[:, 0:128] (ld 256).
__global__ void copy_strided(const float* __restrict__ src, float* __restrict__ dst)
{
    int t = blockIdx.x * blockDim.x + threadIdx.x;
    if (t >= MROWS * DMODEL) return;
    int row = t / DMODEL, col = t % DMODEL;
    dst[(size_t)row * 256 + col] = src[t];
}

// Final 2x64 matvec + clip + polygon update; writes preds[it+1].
__global__ void poly_update(const float* __restrict__ z2, const float* __restrict__ p3w,
                            const float* __restrict__ p3b, float* __restrict__ polys,
                            float* __restrict__ outPred)
{
    int t = blockIdx.x * blockDim.x + threadIdx.x;
    if (t >= MROWS) return;
    const float* z = z2 + (size_t)t * 64;
    float o0 = p3b[0], o1 = p3b[1];
#pragma unroll
    for (int j = 0; j < 64; ++j) { o0 += p3w[j] * z[j]; o1 += p3w[64 + j] * z[j]; }
    o0 = fminf(fmaxf(o0, -CLIP_DIS_V), CLIP_DIS_V);
    o1 = fminf(fmaxf(o1, -CLIP_DIS_V), CLIP_DIS_V);
    float x = polys[t * 2 + 0] + o0, y = polys[t * 2 + 1] + o1;
    x = fminf(fmaxf(x, 0.f), (float)(WW - 1));
    y = fminf(fmaxf(y, 0.f), (float)(HH - 1));
    polys[t * 2 + 0] = x; polys[t * 2 + 1] = y;
    outPred[t * 2 + 0] = x; outPred[t * 2 + 1] = y;
}

// ---------------------------------------------------------------------------
extern "C" void kernel_launch(void* const* d_in, const int* in_sizes, int n_in,
                              void* d_out, int out_size, void* d_ws, size_t ws_size,
                              hipStream_t stream)
{
    const float* cnn_feature = (const float*)d_in[0];
    const float* init_polys  = (const float*)d_in[1];
    const int*   inds        = (const int*)d_in[2];
    const float* conv1_w = (const float*)d_in[3];
    const float* conv1_b = (const float*)d_in[4];
    const float* lin_w   = (const float*)d_in[5];
    const float* lin_b   = (const float*)d_in[6];
    const float* ln_g    = (const float*)d_in[7];
    const float* ln_b    = (const float*)d_in[8];
    const float* q_w     = (const float*)d_in[9];
    const float* q_b     = (const float*)d_in[10];
    const float* k_w     = (const float*)d_in[11];
    const float* k_b     = (const float*)d_in[12];
    const float* v_w     = (const float*)d_in[13];
    const float* v_b     = (const float*)d_in[14];
    const float* in_w    = (const float*)d_in[15];
    const float* in_b    = (const float*)d_in[16];
    const float* out_w   = (const float*)d_in[17];
    const float* out_b   = (const float*)d_in[18];
    const float* f1_w    = (const float*)d_in[19];
    const float* f1_b    = (const float*)d_in[20];
    const float* f2_w    = (const float*)d_in[21];
    const float* f2_b    = (const float*)d_in[22];
    const float* p1_w    = (const float*)d_in[23];
    const float* p1_b    = (const float*)d_in[24];
    const float* p2_w    = (const float*)d_in[25];
    const float* p2_b    = (const float*)d_in[26];
    const float* p3_w    = (const float*)d_in[27];
    const float* p3_b    = (const float*)d_in[28];

    float* out = (float*)d_out;
    float* ws  = (float*)d_ws;

    // Workspace layout (floats)
    size_t off = 0;
    float* polys = ws + off; off += (size_t)MROWS * 2;
    float* stats = ws + off; off += 64;
    float* xbn   = ws + off; off += (size_t)MROWS * CSEG;
    float* hdn   = ws + off; off += (size_t)MROWS * DMODEL;
    float* qn    = ws + off; off += (size_t)MROWS * DMODEL;
    float* A3    = ws + off; off += (size_t)MROWS * 3 * DMODEL;   // q|k|v ; later xcat
    float* B3    = ws + off; off += (size_t)MROWS * 3 * DMODEL;   // qq|kk|vv ; later z1
    float* t1    = ws + off; off += (size_t)MROWS * DMODEL;       // attn out ; later z2
    float* t2    = ws + off; off += (size_t)MROWS * DMODEL;       // post out-proj
    float* ffh   = ws + off; off += (size_t)MROWS * FFDIM;
    if (ws_size < off * sizeof(float)) return;

    const long slab = (long)MROWS * DMODEL;
    const int BIG = 1 << 30;

    auto gemm = [&](const float* X, const float* W, const float* bias,
                    const float* add1, const float* add2, float* Y,
                    int M, int N, int K, int ldY, float scale, int scaleLim,
                    int relu, long xSlab, long ySlab) {
        dim3 grid(N / 64, M / 128);
        dim3 block(32, 4);
        gemm_wmma_f32<<<grid, block, 0, stream>>>(X, W, bias, add1, add2, Y,
                                                  M, N, K, ldY, scale, scaleLim,
                                                  relu, xSlab, ySlab);
    };

    // preds[0] = init_polys ; working polys = init_polys
    copy_flat<<<(MROWS * 2 + 255) / 256, 256, 0, stream>>>(init_polys, out, MROWS * 2);
    copy_flat<<<(MROWS * 2 + 255) / 256, 256, 0, stream>>>(init_polys, polys, MROWS * 2);

    for (int it = 0; it < N_ITERS; ++it) {
        gather_nf<<<(MROWS + 255) / 256, 256, 0, stream>>>(cnn_feature, polys, inds, xbn);
        channel_stats<<<CSEG, 256, 0, stream>>>(xbn, stats);
        bn_apply<<<(MROWS * CSEG + 255) / 256, 256, 0, stream>>>(xbn, stats);

        // hdn = xbn @ lin_w^T + lin_b
        gemm(xbn, lin_w + (size_t)it * DMODEL * CSEG, lin_b + (size_t)it * DMODEL,
             nullptr, nullptr, hdn, MROWS, DMODEL, CSEG, DMODEL, 1.0f, BIG, 0, 0, 0);

        for (int b = 0; b < N_BLOCKS; ++b) {
            const size_t ib  = (size_t)(it * N_BLOCKS + b);
            const float* gln = ln_g + ib * DMODEL;
            const float* bln = ln_b + ib * DMODEL;

            layernorm_rows<<<MROWS / 8, 256, 0, stream>>>(hdn, gln, bln, qn);

            gemm(qn, q_w + ib * DMODEL * DMODEL, q_b + ib * DMODEL, nullptr, nullptr,
                 A3,            MROWS, DMODEL, DMODEL, DMODEL, 1.0f, BIG, 1, 0, 0);
            gemm(qn, k_w + ib * DMODEL * DMODEL, k_b + ib * DMODEL, nullptr, nullptr,
                 A3 + slab,     MROWS, DMODEL, DMODEL, DMODEL, 1.0f, BIG, 1, 0, 0);
            gemm(qn, v_w + ib * DMODEL * DMODEL, v_b + ib * DMODEL, nullptr, nullptr,
                 A3 + 2 * slab, MROWS, DMODEL, DMODEL, DMODEL, 1.0f, BIG, 1, 0, 0);

            // Fused in-projection (N = 384, block-diagonal X over q|k|v slabs)
            const float* inw = in_w + ib * (3 * DMODEL) * DMODEL;
            const float* inb = in_b + ib * (3 * DMODEL);
            const float hscale = 1.0f / sqrtf((float)HDIM);
            gemm(A3, inw, inb, nullptr, nullptr, B3,
                 MROWS, 3 * DMODEL, DMODEL, DMODEL, hscale, DMODEL, 0, slab, slab);

            attention_kernel<<<NPOLY, 192, 0, stream>>>(B3, B3 + slab, B3 + 2 * slab, t1);

            // t2 = t1 @ out_w^T + out_b + hdn
            gemm(t1, out_w + ib * DMODEL * DMODEL, out_b + ib * DMODEL,
                 hdn, nullptr, t2, MROWS, DMODEL, DMODEL, DMODEL, 1.0f, BIG, 0, 0, 0);

            // ffh = relu(t2 @ f1_w^T + f1_b)
            gemm(t2, f1_w + ib * FFDIM * DMODEL, f1_b + ib * FFDIM,
                 nullptr, nullptr, ffh, MROWS, FFDIM, DMODEL, FFDIM, 1.0f, BIG, 1, 0, 0);

            // hdn = (ffh @ f2_w^T + f2_b) + t2 + hdn
            gemm(ffh, f2_w + ib * DMODEL * FFDIM, f2_b + ib * DMODEL,
                 t2, hdn, hdn, MROWS, DMODEL, FFDIM, DMODEL, 1.0f, BIG, 0, 0, 0);
        }

        // ---- prediction head ----
        float* xcat = A3;  // reuse: MROWS x 256
        copy_strided<<<(MROWS * DMODEL + 255) / 256, 256, 0, stream>>>(hdn, xcat);
        gemm(xbn, conv1_w + (size_t)it * DMODEL * CSEG, conv1_b + (size_t)it * DMODEL,
             nullptr, nullptr, xcat + DMODEL, MROWS, DMODEL, CSEG, 256, 1.0f, BIG, 0, 0, 0);

        float* z1 = B3;
        gemm(xcat, p1_w + (size_t)it * DMODEL * 256, p1_b + (size_t)it * DMODEL,
             nullptr, nullptr, z1, MROWS, DMODEL, 256, DMODEL, 1.0f, BIG, 1, 0, 0);
        float* z2 = t1;
        gemm(z1, p2_w + (size_t)it * 64 * DMODEL, p2_b + (size_t)it * 64,
             nullptr, nullptr, z2, MROWS, 64, DMODEL, 64, 1.0f, BIG, 1, 0, 0);

        poly_update<<<(MROWS + 255) / 256, 256, 0, stream>>>(
            z2, p3_w + (size_t)it * 2 * 64, p3_b + (size_t)it * 2,
            polys, out + (size_t)(it + 1) * MROWS * 2);
    }
}